// LCNet_80908593922437
// MI455X (gfx1250) — compile-verified
//
#include <hip/hip_runtime.h>
#include <math.h>

typedef __attribute__((ext_vector_type(2))) float v2f;
typedef __attribute__((ext_vector_type(8))) float v8f;

#define NN 100000

// ---------------- degree / norm precompute ----------------

__global__ __launch_bounds__(256) void k_init_deg(float* __restrict__ deg, int n) {
    int i = blockIdx.x * 256 + threadIdx.x;
    if (i < n) deg[i] = 1.0f;  // self loop contributes 1
}

__global__ __launch_bounds__(256) void k_edge_deg(float* __restrict__ deg,
                                                  const int* __restrict__ dst, int e) {
    int i = blockIdx.x * 256 + threadIdx.x;
    if (i < e) unsafeAtomicAdd(&deg[dst[i]], 1.0f);
}

__global__ __launch_bounds__(256) void k_dinv(float* __restrict__ dinv,
                                              float* __restrict__ selfnorm, int n) {
    int i = blockIdx.x * 256 + threadIdx.x;
    if (i < n) {
        float d = rsqrtf(dinv[i]);   // deg >= 1 always (self loop)
        dinv[i] = d;
        selfnorm[i] = d * d;
    }
}

__global__ __launch_bounds__(256) void k_edge_norm(const float* __restrict__ dinv,
                                                   const int* __restrict__ src,
                                                   const int* __restrict__ dst,
                                                   float* __restrict__ norm, int e) {
    int i = blockIdx.x * 256 + threadIdx.x;
    if (i < e) norm[i] = dinv[src[i]] * dinv[dst[i]];
}

// ---------------- fp32 WMMA GEMM:  H[n x NOUT] = X[n x K] @ W[K x NOUT] ----------------
// One wave per 16-row strip; 16x16 output tiles via V_WMMA_F32_16X16X4_F32.

template <int K, int NOUT>
__global__ __launch_bounds__(256) void k_gemm_wmma(const float* __restrict__ X,
                                                   const float* __restrict__ W,
                                                   float* __restrict__ H, int nrows) {
    constexpr int NT = NOUT / 16;
    const int lane  = threadIdx.x & 31;
    const int wave  = (blockIdx.x * 256 + (int)threadIdx.x) >> 5;
    const int mtile = wave;
    if (mtile * 16 >= nrows) return;  // wave-uniform: EXEC stays all-ones for WMMA

    const int arow = mtile * 16 + (lane & 15);     // A: M = lane%16
    const int kh   = (lane >> 4) << 1;             // A/B: lanes 16-31 hold K+2,K+3
    const int col  = lane & 15;                    // B/C/D: N = lane%16

    v8f acc[NT];
#pragma unroll
    for (int t = 0; t < NT; ++t) acc[t] = (v8f){0.f, 0.f, 0.f, 0.f, 0.f, 0.f, 0.f, 0.f};

    for (int k0 = 0; k0 < K; k0 += 4) {
        v2f a;
        a.x = X[(long long)arow * K + k0 + kh];
        a.y = X[(long long)arow * K + k0 + kh + 1];
#pragma unroll
        for (int t = 0; t < NT; ++t) {
            v2f b;
            b.x = W[(k0 + kh) * NOUT + t * 16 + col];
            b.y = W[(k0 + kh + 1) * NOUT + t * 16 + col];
            acc[t] = __builtin_amdgcn_wmma_f32_16x16x4_f32(
                false, a, false, b, (short)0, acc[t], false, false);
        }
    }

    // C/D layout: VGPR r -> M = r (lanes 0-15) / r+8 (lanes 16-31), N = lane%16
    const int rbase = mtile * 16 + ((lane >> 4) << 3);
#pragma unroll
    for (int t = 0; t < NT; ++t)
#pragma unroll
        for (int r = 0; r < 8; ++r)
            H[(long long)(rbase + r) * NOUT + t * 16 + col] = acc[t][r];
}

// ---------------- aggregation: AGG = selfnorm*H + scatter(norm * H[src] -> dst) ----------------

template <int F>
__global__ __launch_bounds__(256) void k_agg_init(float* __restrict__ agg,
                                                  const float* __restrict__ h,
                                                  const float* __restrict__ selfnorm, int n) {
    constexpr int C = F / 4;
    int tid = blockIdx.x * 256 + threadIdx.x;
    if (tid >= n * C) return;
    int i = tid / C, c = tid % C;
    float s = selfnorm[i];
    float4 hv = ((const float4*)(h + (long long)i * F))[c];
    float4 o;
    o.x = s * hv.x; o.y = s * hv.y; o.z = s * hv.z; o.w = s * hv.w;
    ((float4*)(agg + (long long)i * F))[c] = o;
}

template <int F>
__global__ __launch_bounds__(256) void k_agg_edges(float* __restrict__ agg,
                                                   const float* __restrict__ h,
                                                   const int* __restrict__ src,
                                                   const int* __restrict__ dst,
                                                   const float* __restrict__ norm, int e) {
    constexpr int C = F / 4;  // float4 chunks per row (F=128 -> one wave32 per edge)
    long long tid = (long long)blockIdx.x * 256 + threadIdx.x;
    if (tid >= (long long)e * C) return;
    int ei = (int)(tid / C);
    int c  = (int)(tid % C);
    int s = src[ei], d = dst[ei];
    float w = norm[ei];
    float4 hv = ((const float4*)(h + (long long)s * F))[c];
    float* out = agg + (long long)d * F + c * 4;
    unsafeAtomicAdd(out + 0, w * hv.x);
    unsafeAtomicAdd(out + 1, w * hv.y);
    unsafeAtomicAdd(out + 2, w * hv.z);
    unsafeAtomicAdd(out + 3, w * hv.w);
}

// ---------------- bias + CELU (alpha = 1) ----------------

template <int F>
__global__ __launch_bounds__(256) void k_bias_celu(const float* __restrict__ agg,
                                                   const float* __restrict__ b,
                                                   float* __restrict__ out, int n) {
    int tid = blockIdx.x * 256 + threadIdx.x;
    if (tid >= n * F) return;
    float v = agg[tid] + b[tid % F];
    out[tid] = v > 0.0f ? v : expm1f(v);
}

// ---------------- driver ----------------

static inline int cdiv(long long a, int b) { return (int)((a + b - 1) / b); }

extern "C" void kernel_launch(void* const* d_in, const int* in_sizes, int n_in,
                              void* d_out, int out_size, void* d_ws, size_t ws_size,
                              hipStream_t stream) {
    const float* x  = (const float*)d_in[0];
    const int*  ei  = (const int*)d_in[1];       // [2, E]
    const float* W1 = (const float*)d_in[2];
    const float* b1 = (const float*)d_in[3];
    const float* W2 = (const float*)d_in[4];
    const float* b2 = (const float*)d_in[5];
    const float* W3 = (const float*)d_in[6];
    const float* b3 = (const float*)d_in[7];

    const int n = in_sizes[0] / 64;   // 100000
    const int e = in_sizes[1] / 2;    // 1600000
    const int* src = ei;
    const int* dst = ei + e;

    // workspace layout (floats): dinv[n] | selfnorm[n] | norm[e] | H[n*128] | AGG[n*128]
    float* ws       = (float*)d_ws;
    float* dinv     = ws;
    float* selfnorm = dinv + n;
    float* norm     = selfnorm + n;
    float* H        = norm + e;
    float* AGG      = H + (size_t)n * 128;

    // --- normalization coefficients ---
    k_init_deg<<<cdiv(n, 256), 256, 0, stream>>>(dinv, n);
    k_edge_deg<<<cdiv(e, 256), 256, 0, stream>>>(dinv, dst, e);
    k_dinv<<<cdiv(n, 256), 256, 0, stream>>>(dinv, selfnorm, n);
    k_edge_norm<<<cdiv(e, 256), 256, 0, stream>>>(dinv, src, dst, norm, e);

    const int mtiles = cdiv(n, 16);            // 6250
    const int gemm_blocks = cdiv(mtiles, 8);   // 8 waves/block

    // --- layer 1: 64 -> 128 ---
    k_gemm_wmma<64, 128><<<gemm_blocks, 256, 0, stream>>>(x, W1, H, n);
    k_agg_init<128><<<cdiv((long long)n * 32, 256), 256, 0, stream>>>(AGG, H, selfnorm, n);
    k_agg_edges<128><<<cdiv((long long)e * 32, 256), 256, 0, stream>>>(AGG, H, src, dst, norm, e);
    k_bias_celu<128><<<cdiv((long long)n * 128, 256), 256, 0, stream>>>(AGG, b1, AGG, n);

    // --- layer 2: 128 -> 128 ---
    k_gemm_wmma<128, 128><<<gemm_blocks, 256, 0, stream>>>(AGG, W2, H, n);
    k_agg_init<128><<<cdiv((long long)n * 32, 256), 256, 0, stream>>>(AGG, H, selfnorm, n);
    k_agg_edges<128><<<cdiv((long long)e * 32, 256), 256, 0, stream>>>(AGG, H, src, dst, norm, e);
    k_bias_celu<128><<<cdiv((long long)n * 128, 256), 256, 0, stream>>>(AGG, b2, AGG, n);

    // --- layer 3: 128 -> 64 (CELU applied to final output too) ---
    k_gemm_wmma<128, 64><<<gemm_blocks, 256, 0, stream>>>(AGG, W3, H, n);
    k_agg_init<64><<<cdiv((long long)n * 16, 256), 256, 0, stream>>>(AGG, H, selfnorm, n);
    k_agg_edges<64><<<cdiv((long long)e * 16, 256), 256, 0, stream>>>(AGG, H, src, dst, norm, e);
    k_bias_celu<64><<<cdiv((long long)n * 64, 256), 256, 0, stream>>>(AGG, b3, (float*)d_out, n);
}